// Enc_eeg_35837207118113
// MI455X (gfx1250) — compile-verified
//
#include <hip/hip_runtime.h>
#include <hip/hip_bf16.h>
#include <math.h>

typedef __attribute__((ext_vector_type(2))) float v2f;
typedef __attribute__((ext_vector_type(8))) float v8f;

__device__ __forceinline__ v8f wmma4(v2f a, v2f b, v8f c) {
  // D = A(16x4) * B(4x16) + C, fp32, emits v_wmma_f32_16x16x4_f32
  return __builtin_amdgcn_wmma_f32_16x16x4_f32(false, a, false, b, (short)0, c,
                                               false, false);
}

__device__ __forceinline__ float lrelu02(float v) { return v > 0.f ? v : 0.2f * v; }
__device__ __forceinline__ float eluf(float v)    { return v > 0.f ? v : expm1f(v); }
__device__ __forceinline__ float geluf(float v)   { return 0.5f * v * (1.f + erff(v * 0.70710678118654752f)); }

// ---------------------------------------------------------------------------
// K0: fold conv1(1x26) + avgpool(1,5) + BN1 scale into w_eff[40][30], beta[40]
// ---------------------------------------------------------------------------
__global__ void k_prep(const float* __restrict__ c1w, const float* __restrict__ c1b,
                       const float* __restrict__ g1, const float* __restrict__ b1,
                       const float* __restrict__ m1, const float* __restrict__ v1,
                       float* __restrict__ weff, float* __restrict__ beta) {
  for (int t = threadIdx.x; t < 1200; t += 256) {
    int c = t / 30, j = t - c * 30;
    float s1 = g1[c] * rsqrtf(v1[c] + 1e-5f);
    int tmin = j - 25 > 0 ? j - 25 : 0;
    int tmax = j < 4 ? j : 4;
    float s = 0.f;
    for (int tt = tmin; tt <= tmax; ++tt) s += c1w[c * 26 + (j - tt)];
    weff[t] = 0.2f * s * s1;
  }
  if (threadIdx.x < 40) {
    int c = threadIdx.x;
    float s1 = g1[c] * rsqrtf(v1[c] + 1e-5f);
    beta[c] = (c1b[c] - m1[c]) * s1 + b1[c];
  }
}

// ---------------------------------------------------------------------------
// K1: H(16384x200) = X(16384x200) @ W(200x200)   (fp32 WMMA, 13 N-tiles)
// Padded lanes use clamped (duplicate) columns: never stored, no exec churn.
// ---------------------------------------------------------------------------
__global__ void k_gemm_h(const float* __restrict__ X, const float* __restrict__ W,
                         float* __restrict__ H) {
  const int lane = threadIdx.x & 31;
  const int wave = threadIdx.x >> 5;
  const int gw = blockIdx.x * 8 + wave;        // 0..13311
  const int mt = gw / 13, nt = gw - mt * 13;
  const int lr = lane & 15, lh = lane >> 4;
  const int row = mt * 16 + lr;
  const int col = nt * 16 + lr;
  const int colc = col < 200 ? col : 199;      // clamp: unconditional loads
  v8f acc = {};
  for (int kk = 0; kk < 50; ++kk) {
    int k0 = kk * 4 + 2 * lh;
    v2f a, b;
    a[0] = X[row * 200 + k0];
    a[1] = X[row * 200 + k0 + 1];
    b[0] = W[k0 * 200 + colc];
    b[1] = W[(k0 + 1) * 200 + colc];
    acc = wmma4(a, b, acc);
  }
  if (col < 200) {
    #pragma unroll
    for (int v = 0; v < 8; ++v) H[(mt * 16 + v + 8 * lh) * 200 + col] = acc[v];
  }
}

// ---------------------------------------------------------------------------
// K2: a_s[j] = h[j]·att_src, a_d[j] = h[j]·att_dst, j<1024. One wave per row.
// ---------------------------------------------------------------------------
__global__ void k_scores(const float* __restrict__ H, const float* __restrict__ asrc,
                         const float* __restrict__ adst,
                         float* __restrict__ a_s, float* __restrict__ a_d) {
  const int lane = threadIdx.x & 31;
  const int j = blockIdx.x * 8 + (threadIdx.x >> 5);  // 0..1023
  float s = 0.f, d = 0.f;
  for (int f = lane; f < 200; f += 32) {
    float hv = H[j * 200 + f];
    s += hv * asrc[f];
    d += hv * adst[f];
  }
  for (int o = 16; o > 0; o >>= 1) { s += __shfl_xor(s, o, 32); d += __shfl_xor(d, o, 32); }
  if (lane == 0) { a_s[j] = s; a_d[j] = d; }
}

// ---------------------------------------------------------------------------
// K3a: per target row i: unnormalized softmax weights Wm[i][j] and 1/denom.
// lrelu is monotone -> row max = lrelu(max_j a_s[j] + a_d[i]).
// ---------------------------------------------------------------------------
__global__ void k_attnw(const float* __restrict__ a_s, const float* __restrict__ a_d,
                        float* __restrict__ Wm, float* __restrict__ inv) {
  __shared__ float red[8];
  const int i = blockIdx.x;
  const int tid = threadIdx.x;
  const float adi = a_d[i];
  float m = -1e30f;
  for (int j = tid; j < 1024; j += 256) m = fmaxf(m, a_s[j]);
  for (int o = 16; o > 0; o >>= 1) m = fmaxf(m, __shfl_xor(m, o, 32));
  if ((tid & 31) == 0) red[tid >> 5] = m;
  __syncthreads();
  m = red[0];
  for (int w = 1; w < 8; ++w) m = fmaxf(m, red[w]);
  const float mi = lrelu02(m + adi);
  __syncthreads();
  float ds = 0.f;
  for (int j = tid; j < 1024; j += 256) {
    float wv = expf(lrelu02(a_s[j] + adi) - mi);
    Wm[i * 1024 + j] = wv;
    ds += wv;
  }
  for (int o = 16; o > 0; o >>= 1) ds += __shfl_xor(ds, o, 32);
  if ((tid & 31) == 0) red[tid >> 5] = ds;
  __syncthreads();
  if (tid == 0) {
    float denom = 0.f;
    for (int w = 0; w < 8; ++w) denom += red[w];
    inv[i] = 1.f / denom;
  }
}

// ---------------------------------------------------------------------------
// K3b: XR[0:1024] = X + inv[i] * (Wm @ H[:1024]) + gat_b   (fp32 WMMA)
// M=1024 (64 tiles), N=200 (13 tiles), K=1024. 832 waves -> 104 blocks.
// ---------------------------------------------------------------------------
__global__ void k_attn_gemm(const float* __restrict__ Wm, const float* __restrict__ H,
                            const float* __restrict__ X, const float* __restrict__ inv,
                            const float* __restrict__ gat_b, float* __restrict__ XR) {
  const int lane = threadIdx.x & 31;
  const int wave = threadIdx.x >> 5;
  const int gw = blockIdx.x * 8 + wave;        // 0..831
  const int mt = gw / 13, nt = gw - mt * 13;
  const int lr = lane & 15, lh = lane >> 4;
  const int row = mt * 16 + lr;
  const int col = nt * 16 + lr;
  const int colc = col < 200 ? col : 199;
  v8f acc = {};
  for (int kk = 0; kk < 256; ++kk) {
    int k0 = kk * 4 + 2 * lh;
    v2f a, b;
    a[0] = Wm[row * 1024 + k0];
    a[1] = Wm[row * 1024 + k0 + 1];
    b[0] = H[k0 * 200 + colc];
    b[1] = H[(k0 + 1) * 200 + colc];
    acc = wmma4(a, b, acc);
  }
  if (col < 200) {
    #pragma unroll
    for (int v = 0; v < 8; ++v) {
      int r = mt * 16 + v + 8 * lh;
      XR[r * 200 + col] = X[r * 200 + col] + acc[v] * inv[r] + gat_b[col];
    }
  }
}

// ---------------------------------------------------------------------------
// K3c: rows 1024..16383: x_res = x + h + gat_b (self-loop-only nodes)
// ---------------------------------------------------------------------------
__global__ void k_resid(const float* __restrict__ X, const float* __restrict__ H,
                        const float* __restrict__ gat_b, float* __restrict__ XR) {
  long idx = (long)blockIdx.x * 256 + threadIdx.x;  // < 15360*200
  int f = (int)(idx % 200);
  long r = idx / 200 + 1024;
  XR[r * 200 + f] = X[r * 200 + f] + H[r * 200 + f] + gat_b[f];
}

// ---------------------------------------------------------------------------
// K4: fused conv1eff(30-tap) + elu + conv2 partial GEMM via WMMA.
// grid (16 superchunks, 16 batches), 288 threads = 9 waves (3x3 output tiles).
// Each superchunk = 8 inner chunks of 8 rows; accumulators persist.
// Pad lanes (o>=40 / p>=35) use clamped indices; their C rows/cols unused.
// ---------------------------------------------------------------------------
__global__ void __launch_bounds__(288)
k_convfused(const float* __restrict__ XR, const float* __restrict__ weff,
            const float* __restrict__ beta, const float* __restrict__ W2c,
            float* __restrict__ part) {
  __shared__ float xs[8 * 200];        // 8 input rows
  __shared__ float y1s[320 * 36];      // B operand: K=(c*8+hl) x p (pad 36)
  const int sc = blockIdx.x;           // superchunk 0..15
  const int b  = blockIdx.y;           // batch 0..15
  const int tid = threadIdx.x;
  const int lane = tid & 31, wave = tid >> 5;   // 9 waves
  const int lr = lane & 15, lh = lane >> 4;
  const int ot = wave / 3, pt = wave - ot * 3;  // tile (o,p) in 3x3
  const int o = ot * 16 + lr;
  const int p = pt * 16 + lr;
  const int oc = o < 40 ? o : 39;      // clamped, unconditional loads
  const int pc = p < 35 ? p : 34;
  v8f acc = {};
  for (int ic = 0; ic < 8; ++ic) {
    const int h0 = sc * 64 + ic * 8;
    __syncthreads();                   // prev wmma reads of y1s complete
    for (int t = tid; t < 1600; t += 288)
      xs[t] = XR[((long)b * 1024 + h0 + t / 200) * 200 + (t % 200)];
    __syncthreads();
    for (int t = tid; t < 11200; t += 288) {   // 40c * 8hl * 35p
      int c = t / 280;
      int r = t - c * 280;
      int hl = r / 35;
      int pp = r - hl * 35;
      const float* wrow = &weff[c * 30];
      const float* xrow = &xs[hl * 200 + 5 * pp];
      float s = beta[c];
      #pragma unroll
      for (int jj = 0; jj < 30; ++jj) s += wrow[jj] * xrow[jj];
      y1s[(c * 8 + hl) * 36 + pp] = eluf(s);
    }
    __syncthreads();
    for (int kk = 0; kk < 80; ++kk) {  // K = 40*8 = 320
      int k0 = kk * 4 + 2 * lh;
      int k1 = k0 + 1;
      v2f a, bv;
      a[0]  = W2c[(oc * 40 + (k0 >> 3)) * 1024 + h0 + (k0 & 7)];
      a[1]  = W2c[(oc * 40 + (k1 >> 3)) * 1024 + h0 + (k1 & 7)];
      bv[0] = y1s[k0 * 36 + pc];
      bv[1] = y1s[k1 * 36 + pc];
      acc = wmma4(a, bv, acc);
    }
  }
  #pragma unroll
  for (int v = 0; v < 8; ++v) {
    int oo = ot * 16 + v + 8 * lh;
    int pp = pt * 16 + lr;
    if (oo < 40 && pp < 35)
      part[((long)(b * 16 + sc) * 40 + oo) * 35 + pp] = acc[v];
  }
}

// ---------------------------------------------------------------------------
// K5: reduce 16 partials + BN2 + elu + 1x1 proj -> yflat(16,1400)
// ---------------------------------------------------------------------------
__global__ void k_head(const float* __restrict__ part, const float* __restrict__ c2b,
                       const float* __restrict__ g2, const float* __restrict__ b2,
                       const float* __restrict__ m2, const float* __restrict__ v2,
                       const float* __restrict__ pw, const float* __restrict__ pb,
                       float* __restrict__ yflat) {
  __shared__ float y2s[40 * 35];
  const int b = blockIdx.x, tid = threadIdx.x;
  for (int t = tid; t < 1400; t += 256) {      // t = o*35+p
    int ochn = t / 35, p = t - ochn * 35;
    float s = c2b[ochn];
    for (int sc = 0; sc < 16; ++sc) s += part[((long)(b * 16 + sc) * 40 + ochn) * 35 + p];
    float sb = g2[ochn] * rsqrtf(v2[ochn] + 1e-5f);
    y2s[t] = eluf((s - m2[ochn]) * sb + b2[ochn]);
  }
  __syncthreads();
  for (int t = tid; t < 1400; t += 256) {      // t = p*40+e
    int p = t / 40, e = t - p * 40;
    float s = pb[e];
    for (int ochn = 0; ochn < 40; ++ochn) s += y2s[ochn * 35 + p] * pw[e * 40 + ochn];
    yflat[b * 1400 + t] = s;
  }
}

// ---------------------------------------------------------------------------
// K6: Z(16x1024) = yflat @ W1 + b1 ; G = gelu_exact(Z). M=16 fills one tile.
// ---------------------------------------------------------------------------
__global__ void k_mlp1(const float* __restrict__ Y, const float* __restrict__ W1,
                       const float* __restrict__ B1,
                       float* __restrict__ Z, float* __restrict__ G) {
  const int lane = threadIdx.x & 31;
  const int nt = blockIdx.x * 8 + (threadIdx.x >> 5);  // 0..63
  const int lr = lane & 15, lh = lane >> 4;
  const int col = nt * 16 + lr;
  v8f acc = {};
  for (int kk = 0; kk < 350; ++kk) {
    int k0 = kk * 4 + 2 * lh;
    v2f a, b;
    a[0] = Y[lr * 1400 + k0];
    a[1] = Y[lr * 1400 + k0 + 1];
    b[0] = W1[k0 * 1024 + col];
    b[1] = W1[(k0 + 1) * 1024 + col];
    acc = wmma4(a, b, acc);
  }
  #pragma unroll
  for (int v = 0; v < 8; ++v) {
    int r = v + 8 * lh;
    float z = acc[v] + B1[col];
    Z[r * 1024 + col] = z;
    G[r * 1024 + col] = geluf(z);
  }
}

// ---------------------------------------------------------------------------
// K7: Z2 = Z + G @ W2 + b2
// ---------------------------------------------------------------------------
__global__ void k_mlp2(const float* __restrict__ G, const float* __restrict__ W2,
                       const float* __restrict__ B2, const float* __restrict__ Z,
                       float* __restrict__ Z2) {
  const int lane = threadIdx.x & 31;
  const int nt = blockIdx.x * 8 + (threadIdx.x >> 5);
  const int lr = lane & 15, lh = lane >> 4;
  const int col = nt * 16 + lr;
  v8f acc = {};
  for (int kk = 0; kk < 256; ++kk) {
    int k0 = kk * 4 + 2 * lh;
    v2f a, b;
    a[0] = G[lr * 1024 + k0];
    a[1] = G[lr * 1024 + k0 + 1];
    b[0] = W2[k0 * 1024 + col];
    b[1] = W2[(k0 + 1) * 1024 + col];
    acc = wmma4(a, b, acc);
  }
  #pragma unroll
  for (int v = 0; v < 8; ++v) {
    int r = v + 8 * lh;
    Z2[r * 1024 + col] = Z[r * 1024 + col] + acc[v] + B2[col];
  }
}

// ---------------------------------------------------------------------------
// K8: row LayerNorm -> out(16,1024)
// ---------------------------------------------------------------------------
__global__ void k_ln(const float* __restrict__ Z2, const float* __restrict__ lg,
                     const float* __restrict__ lb, float* __restrict__ out) {
  __shared__ float red[8];
  const int b = blockIdx.x, tid = threadIdx.x;
  float s = 0.f;
  for (int n = tid; n < 1024; n += 256) s += Z2[b * 1024 + n];
  for (int o = 16; o > 0; o >>= 1) s += __shfl_xor(s, o, 32);
  if ((tid & 31) == 0) red[tid >> 5] = s;
  __syncthreads();
  float mu = 0.f;
  for (int w = 0; w < 8; ++w) mu += red[w];
  mu *= (1.f / 1024.f);
  __syncthreads();
  float vs = 0.f;
  for (int n = tid; n < 1024; n += 256) { float d = Z2[b * 1024 + n] - mu; vs += d * d; }
  for (int o = 16; o > 0; o >>= 1) vs += __shfl_xor(vs, o, 32);
  if ((tid & 31) == 0) red[tid >> 5] = vs;
  __syncthreads();
  float var = 0.f;
  for (int w = 0; w < 8; ++w) var += red[w];
  var *= (1.f / 1024.f);
  const float inv = rsqrtf(var + 1e-5f);
  for (int n = tid; n < 1024; n += 256)
    out[b * 1024 + n] = (Z2[b * 1024 + n] - mu) * inv * lg[n] + lb[n];
}

extern "C" void kernel_launch(void* const* d_in, const int* in_sizes, int n_in,
                              void* d_out, int out_size, void* d_ws, size_t ws_size,
                              hipStream_t stream) {
  (void)in_sizes; (void)n_in; (void)out_size; (void)ws_size;
  const float* x     = (const float*)d_in[0];
  const float* gatW  = (const float*)d_in[1];
  const float* attS  = (const float*)d_in[2];
  const float* attD  = (const float*)d_in[3];
  const float* gatB  = (const float*)d_in[4];
  const float* c1w   = (const float*)d_in[5];
  const float* c1b   = (const float*)d_in[6];
  const float* bn1g  = (const float*)d_in[7];
  const float* bn1b  = (const float*)d_in[8];
  const float* bn1m  = (const float*)d_in[9];
  const float* bn1v  = (const float*)d_in[10];
  const float* c2w   = (const float*)d_in[11];
  const float* c2b   = (const float*)d_in[12];
  const float* bn2g  = (const float*)d_in[13];
  const float* bn2b  = (const float*)d_in[14];
  const float* bn2m  = (const float*)d_in[15];
  const float* bn2v  = (const float*)d_in[16];
  const float* pw    = (const float*)d_in[17];
  const float* pb    = (const float*)d_in[18];
  const float* W1    = (const float*)d_in[19];
  const float* b1    = (const float*)d_in[20];
  const float* W2    = (const float*)d_in[21];
  const float* b2    = (const float*)d_in[22];
  const float* lng   = (const float*)d_in[23];
  const float* lnb   = (const float*)d_in[24];

  float* ws   = (float*)d_ws;
  float* H    = ws;                 // 16384*200
  float* XR   = H    + 3276800;     // 16384*200
  float* AS   = XR   + 3276800;     // 1024
  float* AD   = AS   + 1024;        // 1024
  float* WEFF = AD   + 1024;        // 1200
  float* BETA = WEFF + 1200;        // 64
  float* PART = BETA + 64;          // 16*16*1400
  float* YF   = PART + 358400;      // 16*1400
  float* Z    = YF   + 22400;       // 16*1024
  float* G    = Z    + 16384;       // 16*1024
  float* Z2   = G    + 16384;       // 16*1024
  float* WM   = Z2   + 16384;       // 1024*1024 softmax weights
  float* INV  = WM   + 1048576;     // 1024

  k_prep     <<<1, 256, 0, stream>>>(c1w, c1b, bn1g, bn1b, bn1m, bn1v, WEFF, BETA);
  k_gemm_h   <<<1664, 256, 0, stream>>>(x, gatW, H);
  k_scores   <<<128, 256, 0, stream>>>(H, attS, attD, AS, AD);
  k_attnw    <<<1024, 256, 0, stream>>>(AS, AD, WM, INV);
  k_attn_gemm<<<104, 256, 0, stream>>>(WM, H, x, INV, gatB, XR);
  k_resid    <<<12000, 256, 0, stream>>>(x, H, gatB, XR);
  k_convfused<<<dim3(16, 16), 288, 0, stream>>>(XR, WEFF, BETA, c2w, PART);
  k_head     <<<16, 256, 0, stream>>>(PART, c2b, bn2g, bn2b, bn2m, bn2v, pw, pb, YF);
  k_mlp1     <<<8, 256, 0, stream>>>(YF, W1, b1, Z, G);
  k_mlp2     <<<8, 256, 0, stream>>>(G, W2, b2, Z, Z2);
  k_ln       <<<16, 256, 0, stream>>>(Z2, lng, lnb, (float*)d_out);
}